// JPEGBlockingDetector_21629455303278
// MI455X (gfx1250) — compile-verified
//
#include <hip/hip_runtime.h>

// Problem constants (from reference): B=16, C=3, H=W=512, block_size=8.
#define NB   16
#define HH   512
#define WW   512
#define TILE 64

typedef __attribute__((ext_vector_type(2))) float v2f;
typedef __attribute__((ext_vector_type(8))) float v8f;

// d_ws float layout:
//   [0      , 65536 )  colpart [16][8][512]   per-tile-row partial column sums of de_h
//   [65536  , 131072)  rowpart [16][8][512]   per-tile-col partial row sums of de_v
//   [131072 , 139264)  colsum  [16][512]      (index 511 forced to 0 -> WMMA K padding)
//   [139264 , 147456)  rowsum  [16][512]
//   [147456 , 147520)  stats   [64] = bestk_h[16], blocked_h[16], bestk_v[16], blocked_v[16]

// ---------------------------------------------------------------------------
// Kernel 1: fused luminance + cross-diff + tile reduction (the 100 MB pass).
// Grid (8, 8, 16), 256 threads. 64x64 tile with +1 halo row/col in LDS.
// ---------------------------------------------------------------------------
__global__ __launch_bounds__(256) void k_tile_reduce(const float* __restrict__ ref,
                                                     const float* __restrict__ tgt,
                                                     float* __restrict__ colpart,
                                                     float* __restrict__ rowpart) {
  __shared__ float lumR[65][67];   // odd stride -> no LDS bank conflicts on column walk
  __shared__ float lumT[65][67];
  __shared__ float part[4][64];

  const int tid = threadIdx.x;
  const int c0  = blockIdx.x * TILE;
  const int r0  = blockIdx.y * TILE;
  const int b   = blockIdx.z;
  const long plane = (long)HH * WW;
  const float* rb = ref + (long)b * 3 * plane;
  const float* tb = tgt + (long)b * 3 * plane;

  // Load 65x65 luminance tiles (halo out of range -> 0, never used in valid lines)
  for (int i = tid; i < 65 * 65; i += 256) {
    const int lr = i / 65, lc = i % 65;
    const int h = r0 + lr, w = c0 + lc;
    float vR = 0.0f, vT = 0.0f;
    if (h < HH && w < WW) {
      const long o = (long)h * WW + w;
      vR = 0.299f * rb[o] + 0.587f * rb[o + plane] + 0.114f * rb[o + 2 * plane];
      vT = 0.299f * tb[o] + 0.587f * tb[o + plane] + 0.114f * tb[o + 2 * plane];
    }
    lumR[lr][lc] = vR;
    lumT[lr][lc] = vT;
  }
  __syncthreads();

  // ---- column partial sums of de_h = max(|dT_h| - |dR_h|, 0) over 64 tile rows
  {
    const int g = tid >> 6, c = tid & 63;
    float acc = 0.0f;
    for (int r = g * 16; r < g * 16 + 16; ++r) {
      const float dT = fabsf(lumT[r][c] - lumT[r][c + 1]);
      const float dR = fabsf(lumR[r][c] - lumR[r][c + 1]);
      acc += fmaxf(dT - dR, 0.0f);
    }
    part[g][c] = acc;
  }
  __syncthreads();
  if (tid < 64) {
    const float s = part[0][tid] + part[1][tid] + part[2][tid] + part[3][tid];
    colpart[((b * 8 + blockIdx.y) * 512) + c0 + tid] = s;   // each slot written exactly once
  }
  __syncthreads();

  // ---- row partial sums of de_v over 64 tile columns
  {
    const int g = tid >> 6, r = tid & 63;
    float acc = 0.0f;
    for (int c = g * 16; c < g * 16 + 16; ++c) {
      const float dT = fabsf(lumT[r][c] - lumT[r + 1][c]);
      const float dR = fabsf(lumR[r][c] - lumR[r + 1][c]);
      acc += fmaxf(dT - dR, 0.0f);
    }
    part[g][r] = acc;
  }
  __syncthreads();
  if (tid < 64) {
    const float s = part[0][tid] + part[1][tid] + part[2][tid] + part[3][tid];
    rowpart[((b * 8 + blockIdx.x) * 512) + r0 + tid] = s;
  }
}

// ---------------------------------------------------------------------------
// Kernel 2: deterministically combine the 8 tile partials per line.
// line 511 is invalid (n_lines = 511) -> write 0 so it acts as WMMA K-padding.
// ---------------------------------------------------------------------------
__global__ __launch_bounds__(256) void k_combine(const float* __restrict__ colpart,
                                                 const float* __restrict__ rowpart,
                                                 float* __restrict__ colsum,
                                                 float* __restrict__ rowsum) {
  const int t = blockIdx.x * blockDim.x + threadIdx.x;   // 0 .. 16383
  const int dir  = t >> 13;
  const int r    = t & 8191;
  const int b    = r >> 9;
  const int line = r & 511;
  const float* part = dir ? rowpart : colpart;
  float s = 0.0f;
  if (line < 511) {
#pragma unroll
    for (int j = 0; j < 8; ++j) s += part[(b * 8 + j) * 512 + line];
  }
  (dir ? rowsum : colsum)[b * 512 + line] = s;
}

// ---------------------------------------------------------------------------
// Kernel 3: phase statistics via V_WMMA_F32_16X16X4_F32.
// sums[16 batches][8 phases] = L(16x512) x OneHot(512x16), K in 128 steps of 4.
// A layout (ISA 7.12.2, 32-bit 16x4): lane&15 = M, lane>>4 selects K-pair.
// B layout (4x16): lane&15 = N, same K-pair selection. D: VGPR v -> M = v + 8*(lane>>4).
// One wave per block (EXEC all ones), block 0 = columns(H-dir), block 1 = rows(V-dir).
// ---------------------------------------------------------------------------
__global__ __launch_bounds__(32) void k_phase_wmma(const float* __restrict__ colsum,
                                                   const float* __restrict__ rowsum,
                                                   float* __restrict__ stats) {
  __shared__ float S[16][16];
  const float* lines = (blockIdx.x == 0) ? colsum : rowsum;
  float* out = stats + blockIdx.x * 32;

  const int lane = threadIdx.x;
  const int m    = lane & 15;
  const int n    = lane & 15;
  const int koff = (lane >> 4) * 2;

  v8f acc = {};
  for (int t = 0; t < 128; ++t) {
    const int k0 = 4 * t + koff;
    v2f a;
    a.x = lines[m * 512 + k0];
    a.y = lines[m * 512 + k0 + 1];
    v2f bm;
    bm.x = ((k0 & 7) == n) ? 1.0f : 0.0f;
    bm.y = (((k0 + 1) & 7) == n) ? 1.0f : 0.0f;
    // (neg_a, A, neg_b, B, c_mod, C, reuse_a, reuse_b)
    acc = __builtin_amdgcn_wmma_f32_16x16x4_f32(false, a, false, bm, (short)0, acc,
                                                false, false);
  }

  const int mg = (lane >> 4) * 8;
#pragma unroll
  for (int v = 0; v < 8; ++v) S[mg + v][lane & 15] = acc[v];
  __syncthreads();

  if (lane < 16) {                       // lane == batch index
    float sums[8];
    float total = 0.0f;
#pragma unroll
    for (int k = 0; k < 8; ++k) { sums[k] = S[lane][k]; total += sums[k]; }
    float best = -1.0f; int bk = 0;
#pragma unroll
    for (int k = 0; k < 8; ++k) {        // counts: 64 for k<7, 63 for k==7 (511 lines)
      const float cnt   = (k < 7) ? 64.0f : 63.0f;
      const float a_k   = sums[k] / cnt;
      const float bg    = (total - sums[k]) / (511.0f - cnt);
      const float ratio = a_k / (bg + 1e-8f);
      if (ratio > best) { best = ratio; bk = k; }   // first-max, like jnp.argmax
    }
    out[lane]      = (float)bk;
    out[16 + lane] = (best > (1.0f / 0.35f)) ? 1.0f : 0.0f;
  }
}

// ---------------------------------------------------------------------------
// Kernel 4: broadcast the winning phases to the (16,1,512,512) binary mask.
// float4 stores -> 16.8 MB streaming write.
// ---------------------------------------------------------------------------
__global__ __launch_bounds__(256) void k_write_mask(const float* __restrict__ stats,
                                                    float* __restrict__ out) {
  const int t  = blockIdx.x * blockDim.x + threadIdx.x;   // one float4 per thread
  const long e = (long)t * 4;
  const int b   = (int)(e >> 18);          // 512*512 = 2^18 per batch
  const int rem = (int)(e & 262143);
  const int h   = rem >> 9;
  const int w0  = rem & 511;

  const float bkh = stats[b],      blh = stats[16 + b];
  const float bkv = stats[32 + b], blv = stats[48 + b];
  const bool rowhit = (blv > 0.5f) && ((h & 7) == (int)bkv) && (h < 511);

  float4 r;
  float* rp = &r.x;
#pragma unroll
  for (int j = 0; j < 4; ++j) {
    const int w = w0 + j;
    const bool colhit = (blh > 0.5f) && ((w & 7) == (int)bkh) && (w < 511);
    rp[j] = (rowhit || colhit) ? 1.0f : 0.0f;
  }
  reinterpret_cast<float4*>(out)[t] = r;
}

// ---------------------------------------------------------------------------
extern "C" void kernel_launch(void* const* d_in, const int* in_sizes, int n_in,
                              void* d_out, int out_size, void* d_ws, size_t ws_size,
                              hipStream_t stream) {
  const float* ref = (const float*)d_in[0];
  const float* tgt = (const float*)d_in[1];
  float* ws = (float*)d_ws;

  float* colpart = ws;              // 16*8*512
  float* rowpart = ws + 65536;      // 16*8*512
  float* colsum  = ws + 131072;     // 16*512
  float* rowsum  = ws + 139264;     // 16*512
  float* stats   = ws + 147456;     // 64 floats

  k_tile_reduce<<<dim3(8, 8, NB), 256, 0, stream>>>(ref, tgt, colpart, rowpart);
  k_combine   <<<dim3(64),        256, 0, stream>>>(colpart, rowpart, colsum, rowsum);
  k_phase_wmma<<<dim3(2),          32, 0, stream>>>(colsum, rowsum, stats);
  k_write_mask<<<dim3(4096),      256, 0, stream>>>(stats, (float*)d_out);
}